// PairProductSpin_17532056502831
// MI455X (gfx1250) — compile-verified
//
#include <hip/hip_runtime.h>

typedef __attribute__((ext_vector_type(2))) float v2f;
typedef __attribute__((ext_vector_type(8))) float v8f;

#define N_DIM    256
#define HALF_DIM 128
#define LDA      129    // pad to avoid LDS bank conflicts on column walks
#define NB       16
#define NTHREADS 256

__global__ __launch_bounds__(NTHREADS)
void ppspin_det_kernel(const float* __restrict__ x,
                       const float* __restrict__ F,
                       float* __restrict__ out)
{
    __shared__ float A[HALF_DIM * LDA];   // 128x128 submatrix, padded rows
    __shared__ float xsh[N_DIM];
    __shared__ float red_val[HALF_DIM];
    __shared__ int   red_idx[HALF_DIM];
    __shared__ int   iu[HALF_DIM];
    __shared__ int   idn[HALF_DIM];

    const int tid = threadIdx.x;
    const int b   = blockIdx.x;

    // ---- load x row, build row/col selection lists ----
    for (int i = tid; i < N_DIM; i += NTHREADS)
        xsh[i] = x[(size_t)b * N_DIM + i];
    __syncthreads();
    if (tid == 0) {
        int nu = 0, nd = 0;
        for (int i = 0; i < N_DIM; ++i) {
            if (xsh[i] > 0.0f) iu[nu++] = i;
            else               idn[nd++] = i;
        }
    }
    __syncthreads();

    // ---- gather 128x128 submatrix from F (analytic off-diagonal index) ----
    for (int e = tid; e < HALF_DIM * HALF_DIM; e += NTHREADS) {
        const int i = e >> 7, j = e & (HALF_DIM - 1);
        const int r = iu[i], c = idn[j];              // r != c guaranteed
        const int idx = r * (N_DIM - 1) + c - (c > r ? 1 : 0);
        A[i * LDA + j] = F[idx];
    }
    __syncthreads();

    float det = 1.0f;   // accumulated redundantly by all threads (uniform values)

    const int wave  = tid >> 5;
    const int lane  = tid & 31;
    const int col   = lane & 15;
    const int mrow  = lane & 15;          // A-fragment row
    const int rhalf = (lane >> 4) << 3;   // C rows: 0 or +8
    const int khalf = (lane >> 4) << 1;   // A/B K:  0 or +2

    for (int k = 0; k < HALF_DIM / NB; ++k) {
        const int k0 = k * NB;

        // ---- panel factorization (columns k0..k0+15), partial pivoting ----
        for (int c = k0; c < k0 + NB; ++c) {
            const int nrows = HALF_DIM - c;
            if (tid < HALF_DIM) {
                float v = 0.0f; int ix = c;
                if (tid < nrows) { v = fabsf(A[(c + tid) * LDA + c]); ix = c + tid; }
                red_val[tid] = v; red_idx[tid] = ix;
            }
            __syncthreads();
            for (int s = 64; s > 0; s >>= 1) {
                if (tid < s) {
                    if (red_val[tid + s] > red_val[tid]) {
                        red_val[tid] = red_val[tid + s];
                        red_idx[tid] = red_idx[tid + s];
                    }
                }
                __syncthreads();
            }
            const int piv = red_idx[0];
            __syncthreads();
            if (piv != c) {
                if (tid < HALF_DIM) {       // full row swap (L and trailing)
                    const float t0 = A[c * LDA + tid];
                    A[c * LDA + tid]   = A[piv * LDA + tid];
                    A[piv * LDA + tid] = t0;
                }
                det = -det;
            }
            __syncthreads();
            const float d = A[c * LDA + c];
            det *= d;
            const float rinv = 1.0f / d;
            const int r = c + 1 + tid;
            if (r < HALF_DIM) {             // multipliers + in-panel rank-1 update
                const float m = A[r * LDA + c] * rinv;
                A[r * LDA + c] = m;
                for (int j = c + 1; j < k0 + NB; ++j)
                    A[r * LDA + j] -= m * A[c * LDA + j];
            }
            __syncthreads();
        }

        // ---- triangular solve: U12 = L11^{-1} A12 (forward substitution) ----
        for (int i = 1; i < NB; ++i) {
            const int r = k0 + i;
            for (int j = k0 + NB + tid; j < HALF_DIM; j += NTHREADS) {
                float s = 0.0f;
                for (int cc = k0; cc < r; ++cc)
                    s += A[r * LDA + cc] * A[cc * LDA + j];
                A[r * LDA + j] -= s;
            }
            __syncthreads();
        }

        // ---- trailing update A22 -= L21 * U12 via f32 WMMA 16x16x4 ----
        // 1x2 column-tile register blocking per wave: A-fragment (L21) reused
        // across both tiles, two independent accumulator chains interleave to
        // hide the WMMA->WMMA RAW hazard latency.
        const int trail = HALF_DIM - k0 - NB;
        const int nt    = trail >> 4;          // 16x16 tiles per side
        const int ncolp = (nt + 1) >> 1;       // column-pair slots per tile row
        for (int t = wave; t < nt * ncolp; t += (NTHREADS >> 5)) {
            const int ti  = t / ncolp;
            const int tj0 = (t - ti * ncolp) << 1;
            const int r0  = k0 + NB + ti * 16;
            const int c0  = k0 + NB + tj0 * 16;
            const bool two = (tj0 + 1) < nt;   // wave-uniform

            if (two) {
                const int c1 = c0 + 16;
                v8f acc0, acc1;
                #pragma unroll
                for (int v = 0; v < 8; ++v) {
                    acc0[v] = A[(r0 + rhalf + v) * LDA + c0 + col];
                    acc1[v] = A[(r0 + rhalf + v) * LDA + c1 + col];
                }
                #pragma unroll
                for (int kk = 0; kk < NB; kk += 4) {
                    v2f af, bf0, bf1;
                    af[0] = -A[(r0 + mrow) * LDA + (k0 + kk + khalf + 0)];
                    af[1] = -A[(r0 + mrow) * LDA + (k0 + kk + khalf + 1)];
                    bf0[0] = A[(k0 + kk + khalf + 0) * LDA + c0 + col];
                    bf0[1] = A[(k0 + kk + khalf + 1) * LDA + c0 + col];
                    bf1[0] = A[(k0 + kk + khalf + 0) * LDA + c1 + col];
                    bf1[1] = A[(k0 + kk + khalf + 1) * LDA + c1 + col];
                    acc0 = __builtin_amdgcn_wmma_f32_16x16x4_f32(
                        false, af, false, bf0, (short)0, acc0, false, false);
                    acc1 = __builtin_amdgcn_wmma_f32_16x16x4_f32(
                        false, af, false, bf1, (short)0, acc1, false, false);
                }
                #pragma unroll
                for (int v = 0; v < 8; ++v) {
                    A[(r0 + rhalf + v) * LDA + c0 + col] = acc0[v];
                    A[(r0 + rhalf + v) * LDA + c1 + col] = acc1[v];
                }
            } else {
                v8f acc0;
                #pragma unroll
                for (int v = 0; v < 8; ++v)
                    acc0[v] = A[(r0 + rhalf + v) * LDA + c0 + col];
                #pragma unroll
                for (int kk = 0; kk < NB; kk += 4) {
                    v2f af, bf0;
                    af[0] = -A[(r0 + mrow) * LDA + (k0 + kk + khalf + 0)];
                    af[1] = -A[(r0 + mrow) * LDA + (k0 + kk + khalf + 1)];
                    bf0[0] = A[(k0 + kk + khalf + 0) * LDA + c0 + col];
                    bf0[1] = A[(k0 + kk + khalf + 1) * LDA + c0 + col];
                    acc0 = __builtin_amdgcn_wmma_f32_16x16x4_f32(
                        false, af, false, bf0, (short)0, acc0, false, false);
                }
                #pragma unroll
                for (int v = 0; v < 8; ++v)
                    A[(r0 + rhalf + v) * LDA + c0 + col] = acc0[v];
            }
        }
        __syncthreads();
    }

    if (tid == 0) out[b] = det;
}

extern "C" void kernel_launch(void* const* d_in, const int* in_sizes, int n_in,
                              void* d_out, int out_size, void* d_ws, size_t ws_size,
                              hipStream_t stream) {
    const float* x = (const float*)d_in[0];   // (B, 256) fp32
    const float* F = (const float*)d_in[1];   // (65280,) fp32
    // d_in[2] (index) is reproduced analytically in-kernel
    float* out = (float*)d_out;               // (B,) fp32

    dim3 grid(out_size), block(NTHREADS);
    ppspin_det_kernel<<<grid, block, 0, stream>>>(x, F, out);
}